// GCNBlock_58566174048907
// MI455X (gfx1250) — compile-verified
//
#include <hip/hip_runtime.h>
#include <hip/hip_bf16.h>

typedef float v2f __attribute__((ext_vector_type(2)));
typedef float v8f __attribute__((ext_vector_type(8)));

#define DIM 64
#define KP  32          // number of k-pairs (DIM/2)
#define BP  80          // padded v2f stride per k-pair row (64 + 16 pad -> bank-split halves)

// ---------------------------------------------------------------------------
// deg[n] = 1.0 (self loop)
__global__ void __launch_bounds__(256) k_deg_init(float* __restrict__ deg, int n) {
    int i = blockIdx.x * blockDim.x + threadIdx.x;
    if (i < n) deg[i] = 1.0f;
}

// deg[col[e]] += 1
__global__ void __launch_bounds__(256) k_deg_accum(const int* __restrict__ col,
                                                   float* __restrict__ deg, int e) {
    int i = blockIdx.x * blockDim.x + threadIdx.x;
    if (i < e) atomicAdd(&deg[col[i]], 1.0f);
}

// dinv[n] = rsqrt(deg[n])   (deg >= 1 always, self loop)
__global__ void __launch_bounds__(256) k_dinv(const float* __restrict__ deg,
                                              float* __restrict__ dinv, int n) {
    int i = blockIdx.x * blockDim.x + threadIdx.x;
    if (i < n) dinv[i] = rsqrtf(deg[i]);
}

// norm[e] = dinv[row[e]] * dinv[col[e]]
__global__ void __launch_bounds__(256) k_norm(const int* __restrict__ row,
                                              const int* __restrict__ col,
                                              const float* __restrict__ dinv,
                                              float* __restrict__ norm, int e) {
    int i = blockIdx.x * blockDim.x + threadIdx.x;
    if (i < e) norm[i] = dinv[row[i]] * dinv[col[i]];
}

// ---------------------------------------------------------------------------
// Fused GEMM + self-loop/bias epilogue using V_WMMA_F32_16X16X4_F32.
//   Y       = X @ W                         (raw product, consumed by edge gather)
//   OutInit = (X @ W) * dinv[row]^2 + bias  (initializes the scatter accumulator)
//
// W staged in LDS in K-pair-interleaved form: sB[k2*BP + col] = {W[2k2][col],
// W[2k2+1][col]} so every WMMA B fragment is ONE aligned ds_load_b64.
// BP=80 pads consecutive k2 rows by 128 B so the two lane-halves of a wave hit
// disjoint LDS bank halves (0-31 vs 32-63): conflict-free b64 loads.
//
// fp32 WMMA VGPR layouts (ISA 7.12.2):
//   A 16x4 : lane&15 = M row; lane>>4 selects K{0,1} vs K{2,3}; v2f = {K, K+1}
//   B 4x16 : lane&15 = N col; lane>>4 selects K rows {0,1} vs {2,3}; v2f = {kr, kr+1}
//   C 16x16: vgpr j, lanes 0-15 -> M=j, lanes 16-31 -> M=j+8; N = lane&15
__global__ void __launch_bounds__(256) k_gemm_wmma_fused(const float* __restrict__ X,
                                                         const float* __restrict__ W,
                                                         const float* __restrict__ dinv,
                                                         const float* __restrict__ bias,
                                                         float* __restrict__ Y,
                                                         float* __restrict__ OutInit,
                                                         int nrows) {
    __shared__ v2f sB[KP * BP];          // 32 k-pair rows x 80 v2f = 20 KB
    {   // cooperative stage+interleave: thread t -> W row r = t>>2, 16 cols
        int t  = threadIdx.x;
        int r  = t >> 2;                 // 0..63
        int cb = (t & 3) * 16;           // 0,16,32,48
        int k2 = r >> 1;
        int kb = r & 1;
        const float4* src = (const float4*)(W + r * DIM + cb);
        float* dst = (float*)sB + (k2 * BP) * 2 + kb;
        #pragma unroll
        for (int i = 0; i < 4; ++i) {
            float4 v = src[i];
            int c = cb + i * 4;
            dst[(c + 0) * 2] = v.x;
            dst[(c + 1) * 2] = v.y;
            dst[(c + 2) * 2] = v.z;
            dst[(c + 3) * 2] = v.w;
        }
    }
    __syncthreads();

    int wave = (blockIdx.x * blockDim.x + threadIdx.x) >> 5;
    int lane = threadIdx.x & 31;
    int m0 = wave * 16;
    if (m0 >= nrows) return;

    int half = lane >> 4;     // 0: lanes 0-15, 1: lanes 16-31
    int l15  = lane & 15;

    v8f acc0 = {}, acc1 = {}, acc2 = {}, acc3 = {};
    const float* xrow = X + (size_t)(m0 + l15) * DIM;

    #pragma unroll 4
    for (int k = 0; k < DIM; k += 4) {
        int kk = k + half * 2;                       // even -> 8B-aligned v2f load
        v2f a = *(const v2f*)(xrow + kk);

        const v2f* bp = sB + ((k >> 1) + half) * BP + l15;
        v2f b0 = bp[0];
        v2f b1 = bp[16];
        v2f b2 = bp[32];
        v2f b3 = bp[48];

        acc0 = __builtin_amdgcn_wmma_f32_16x16x4_f32(false, a, false, b0, (short)0, acc0, false, false);
        acc1 = __builtin_amdgcn_wmma_f32_16x16x4_f32(false, a, false, b1, (short)0, acc1, false, false);
        acc2 = __builtin_amdgcn_wmma_f32_16x16x4_f32(false, a, false, b2, (short)0, acc2, false, false);
        acc3 = __builtin_amdgcn_wmma_f32_16x16x4_f32(false, a, false, b3, (short)0, acc3, false, false);
    }

    // Epilogue: per-column bias (column = nblock*16 + l15), per-row dinv^2.
    float bv0 = bias[l15];
    float bv1 = bias[16 + l15];
    float bv2 = bias[32 + l15];
    float bv3 = bias[48 + l15];

    size_t base = (size_t)(m0 + half * 8) * DIM + l15;
    float*       yb = Y + base;
    float*       ob = OutInit + base;
    const float* dv = dinv + m0 + half * 8;

    #pragma unroll
    for (int j = 0; j < 8; ++j) {
        float di = dv[j];
        float w  = di * di;
        float r0 = acc0[j], r1 = acc1[j], r2 = acc2[j], r3 = acc3[j];
        yb[j * DIM + 0]  = r0;
        yb[j * DIM + 16] = r1;
        yb[j * DIM + 32] = r2;
        yb[j * DIM + 48] = r3;
        ob[j * DIM + 0]  = r0 * w + bv0;
        ob[j * DIM + 16] = r1 * w + bv1;
        ob[j * DIM + 32] = r2 * w + bv2;
        ob[j * DIM + 48] = r3 * w + bv3;
    }
}

// ---------------------------------------------------------------------------
// out[col[e]][k] += xw[row[e]][k] * norm[e]   -- 16 threads/edge, float4 gather,
// 4 fp32 global atomics per thread (destination is L2-resident: 12.8 MB << 192 MB).
__global__ void __launch_bounds__(256) k_agg_edges(const int* __restrict__ row,
                                                   const int* __restrict__ col,
                                                   const float* __restrict__ norm,
                                                   const float* __restrict__ xw,
                                                   float* __restrict__ out, int e) {
    int tid = blockIdx.x * blockDim.x + threadIdx.x;
    int edge = tid >> 4;
    int q    = (tid & 15) * 4;
    if (edge >= e) return;
    int r = row[edge];
    int c = col[edge];
    float w = norm[edge];
    const float4 v = *(const float4*)(xw + (size_t)r * DIM + q);
    float* dst = out + (size_t)c * DIM + q;
    atomicAdd(dst + 0, v.x * w);
    atomicAdd(dst + 1, v.y * w);
    atomicAdd(dst + 2, v.z * w);
    atomicAdd(dst + 3, v.w * w);
}

// ---------------------------------------------------------------------------
extern "C" void kernel_launch(void* const* d_in, const int* in_sizes, int n_in,
                              void* d_out, int out_size, void* d_ws, size_t ws_size,
                              hipStream_t stream) {
    const float* x  = (const float*)d_in[0];
    const int*   ei = (const int*)d_in[1];
    const float* W1 = (const float*)d_in[2];
    const float* b1 = (const float*)d_in[3];
    const float* W2 = (const float*)d_in[4];
    const float* b2 = (const float*)d_in[5];

    const int N = in_sizes[0] / DIM;      // 50000
    const int E = in_sizes[1] / 2;        // 800000
    const int* row = ei;                  // edge_index[0]
    const int* col = ei + E;              // edge_index[1]

    float* ws   = (float*)d_ws;
    float* deg  = ws;                     // N
    float* dinv = deg  + N;               // N
    float* nrm  = dinv + N;               // E
    float* bufA = nrm  + E;               // N*64  (x@W1, then h1@W2)
    float* bufB = bufA + (size_t)N * DIM; // N*64  (h1)
    float* outp = (float*)d_out;          // N*64

    const int B = 256;

    // --- normalization (computed once, shared by both layers) ---
    k_deg_init <<<(N + B - 1) / B, B, 0, stream>>>(deg, N);
    k_deg_accum<<<(E + B - 1) / B, B, 0, stream>>>(col, deg, E);
    k_dinv     <<<(N + B - 1) / B, B, 0, stream>>>(deg, dinv, N);
    k_norm     <<<(E + B - 1) / B, B, 0, stream>>>(row, col, dinv, nrm, E);

    const int gemm_waves  = N / 16;                       // 3125 tiles of 16 rows
    const int gemm_blocks = (gemm_waves * 32 + B - 1) / B;
    const int edge_blocks = ((size_t)E * 16 + B - 1) / B;

    // --- layer 1: h1 = A_hat (x @ W1) + b1 ---
    k_gemm_wmma_fused<<<gemm_blocks, B, 0, stream>>>(x, W1, dinv, b1, bufA, bufB, N);
    k_agg_edges      <<<edge_blocks, B, 0, stream>>>(row, col, nrm, bufA, bufB, E);

    // --- layer 2: out = A_hat (h1 @ W2) + b2 ---
    k_gemm_wmma_fused<<<gemm_blocks, B, 0, stream>>>(bufB, W2, dinv, b2, bufA, outp, N);
    k_agg_edges      <<<edge_blocks, B, 0, stream>>>(row, col, nrm, bufA, outp, E);
}